// DecodeSSDPredictions_10436770529839
// MI455X (gfx1250) — compile-verified
//
#include <hip/hip_runtime.h>
#include <stdint.h>

// ---------------- problem constants (match reference) ----------------
#define IMG_SCALE   512.0f
#define N_CLASSES   81
#define ROW         93          // 81 cls + 4 off + 2 cxcy + 2 wh + 4 var
#define N_BOXES_K   24564
#define BATCH       32
#define CONF_THR    0.5f
#define IOU_THR     0.35f
#define NMS_MAX_K   100
#define NUM_PRED_K  10

// candidate workspace layout (per batch item, SoA):
//   score[CAP] x0[CAP] y0[CAP] x1[CAP] y1[CAP] area[CAP] origidx[CAP] cls[CAP]
// First 7 fields are staged into LDS by the TDM; cls stays in global.
#define CAP          8192
#define FIELDS       8
#define STAGE_FIELDS 7
#define STAGE_ELEMS  (STAGE_FIELDS * CAP)   // 57344 f32 = 224KB LDS, tile_dim0 < 65536

// ---------------- TDM (Tensor Data Mover) helpers ----------------
typedef uint32_t u32x4_t __attribute__((ext_vector_type(4)));
typedef int      i32x4_t __attribute__((ext_vector_type(4)));
typedef int      i32x8_t __attribute__((ext_vector_type(8)));

#if defined(__has_builtin)
#if __has_builtin(__builtin_amdgcn_tensor_load_to_lds)
#define HAVE_TDM 1
#endif
#endif

// 1-D contiguous tile: copy nElem f32 from global -> LDS via the TDM.
// D# per CDNA5 ISA ch.8: group0 = {count/lds_addr/global_addr/type},
// group1 = {data_size, tensor_dim0, tensor_dim1=1, tile_dim0, stride}.
__device__ __forceinline__ void tdm_load_1d_f32(void* lds_dst, const void* gsrc,
                                                uint32_t nElem) {
#if defined(HAVE_TDM)
  uint64_t ga = (uint64_t)(uintptr_t)gsrc;
  uint32_t la = (uint32_t)(uintptr_t)lds_dst;     // low 32 bits = LDS byte offset
  u32x4_t g0;
  g0.x = 1u;                                       // count=1 valid descriptor
  g0.y = la;                                       // lds_addr
  g0.z = (uint32_t)ga;                             // global_addr[31:0]
  g0.w = (uint32_t)((ga >> 32) & 0x01FFFFFFu) | (2u << 30);  // addr[56:32] | type=2
  i32x8_t g1;
  g1[0] = (int)(2u << 16);                         // workgroup_mask=0, data_size=2 (4B)
  g1[1] = (int)((nElem & 0xFFFFu) << 16);          // tensor_dim0 lo16
  g1[2] = (int)(((nElem >> 16) & 0xFFFFu) | (1u << 16)); // tensor_dim0 hi16 | tensor_dim1=1
  g1[3] = (int)((nElem & 0xFFFFu) << 16);          // tile_dim0 (16-bit)
  g1[4] = 0;                                       // tile_dim1=0 (unused), tile_dim2=0
  g1[5] = (int)nElem;                              // tensor_dim0_stride lo32
  g1[6] = 0;                                       // stride hi16 | dim1_stride lo16
  g1[7] = 0;
  i32x4_t gz4 = {0, 0, 0, 0};
#if defined(__clang_major__) && (__clang_major__ >= 23)
  i32x8_t gz8 = {0, 0, 0, 0, 0, 0, 0, 0};
  __builtin_amdgcn_tensor_load_to_lds(g0, g1, gz4, gz4, gz8, 0);
#else
  __builtin_amdgcn_tensor_load_to_lds(g0, g1, gz4, gz4, 0);
#endif
  __builtin_amdgcn_s_wait_tensorcnt(0);
#else
  (void)lds_dst; (void)gsrc; (void)nElem;
#endif
}

// ---------------- Kernel 1: decode + argmax + compaction ----------------
// One wave32 per box. Lanes read classes {l, l+32, l+64}; butterfly argmax
// (lowest-index tie-break = jnp.argmax semantics); lane 0 decodes + appends.
__global__ __launch_bounds__(256)
void ssd_decode_kernel(const float* __restrict__ y,
                       float* __restrict__ cand,
                       int* __restrict__ counters) {
  const int wave = (int)((blockIdx.x << 3) | (threadIdx.x >> 5));
  const int lane = (int)(threadIdx.x & 31);
  if (wave >= BATCH * N_BOXES_K) return;

  const float* __restrict__ p = y + (size_t)wave * ROW;
  // warm the L2 stream ahead of us (gfx1250 global_prefetch_b8)
  __builtin_prefetch(p + 8 * ROW, 0, 1);

  float bv = -1.0f;
  int   bc = 0;
#pragma unroll
  for (int k = 0; k < 3; ++k) {
    int c = lane + (k << 5);
    if (c < N_CLASSES) {
      float v = p[c];
      if (v > bv) { bv = v; bc = c; }     // strict > keeps first max per lane
    }
  }
#pragma unroll
  for (int off = 16; off > 0; off >>= 1) { // wave32 butterfly
    float ov = __shfl_xor(bv, off, 32);
    int   oc = __shfl_xor(bc, off, 32);
    if (ov > bv || (ov == bv && oc < bc)) { bv = ov; bc = oc; }
  }

  if (lane == 0 && bc != 0 && bv >= CONF_THR) {
    const int b = wave / N_BOXES_K;
    const int n = wave - b * N_BOXES_K;   // original box index within this image
    const float o0 = p[81], o1 = p[82], o2 = p[83], o3 = p[84];
    const float pcx = p[85], pcy = p[86], pw = p[87], ph = p[88];
    const float v0 = p[89], v1 = p[90], v2 = p[91], v3 = p[92];
    const float cx = o0 * v0 * pw + pcx;
    const float cy = o1 * v1 * ph + pcy;
    const float w  = __expf(o2 * v2) * pw;
    const float h  = __expf(o3 * v3) * ph;
    const float x0 = (cx - 0.5f * w) * IMG_SCALE;
    const float y0 = (cy - 0.5f * h) * IMG_SCALE;
    const float x1 = (cx + 0.5f * w) * IMG_SCALE;
    const float y1 = (cy + 0.5f * h) * IMG_SCALE;
    const int slot = atomicAdd(&counters[b], 1);
    if (slot < CAP) {
      float* base = cand + (size_t)b * (FIELDS * CAP);
      base[0 * CAP + slot] = bv;
      base[1 * CAP + slot] = x0;
      base[2 * CAP + slot] = y0;
      base[3 * CAP + slot] = x1;
      base[4 * CAP + slot] = y1;
      base[5 * CAP + slot] = (x1 - x0) * (y1 - y0);
      base[6 * CAP + slot] = (float)n;    // original index (exact in f32: < 2^24)
      base[7 * CAP + slot] = (float)bc;
    }
  }
}

// ---------------- Kernel 2: greedy NMS in LDS (one block per image) ------
// 320KB LDS/WGP lets the whole candidate table live on-chip.
__global__ __launch_bounds__(256)
void ssd_nms_kernel(const float* __restrict__ cand,
                    const int* __restrict__ counters,
                    float* __restrict__ out) {
  __shared__ float smem[STAGE_ELEMS];      // 224KB SoA: sc,x0,y0,x1,y1,area,oidx
  __shared__ float red_val[256];
  __shared__ int   red_idx[256];
  __shared__ int   red_oid[256];
  __shared__ float kept_sc[NMS_MAX_K];
  __shared__ int   kept_id[NMS_MAX_K];
  __shared__ float bbox[5];
  __shared__ int   ok_flag;

  const int b   = (int)blockIdx.x;
  const int tid = (int)threadIdx.x;
  const float* __restrict__ gbase = cand + (size_t)b * (FIELDS * CAP);
  int cnt = counters[b];
  if (cnt > CAP) cnt = CAP;

#if defined(HAVE_TDM)
  if (tid < 32) {                           // wave 0 drives the DMA engine
    tdm_load_1d_f32((void*)smem, (const void*)gbase, (uint32_t)STAGE_ELEMS);
  }
#else
  for (int i = tid; i < STAGE_ELEMS; i += 256) smem[i] = gbase[i];
#endif
  __syncthreads();

  float* __restrict__ sc   = smem + 0 * CAP;
  float* __restrict__ x0a  = smem + 1 * CAP;
  float* __restrict__ y0a  = smem + 2 * CAP;
  float* __restrict__ x1a  = smem + 3 * CAP;
  float* __restrict__ y1a  = smem + 4 * CAP;
  float* __restrict__ ara  = smem + 5 * CAP;
  float* __restrict__ oixa = smem + 6 * CAP;

  const float NEG_INF = -__builtin_inff();

  for (int it = 0; it < NMS_MAX_K; ++it) {
    // block-wide argmax over alive scores; ties -> smallest ORIGINAL box index
    float bvv = NEG_INF;
    int   bii = -1;
    int   boi = 0x7FFFFFFF;
    for (int i = tid; i < cnt; i += 256) {
      float v  = sc[i];
      int   oi = (int)oixa[i];
      if (v > bvv || (v == bvv && oi < boi)) { bvv = v; bii = i; boi = oi; }
    }
    red_val[tid] = bvv;
    red_idx[tid] = bii;
    red_oid[tid] = boi;
    __syncthreads();
#pragma unroll
    for (int s = 128; s > 0; s >>= 1) {
      if (tid < s) {
        float ov = red_val[tid + s];
        int   oi = red_idx[tid + s];
        int   oo = red_oid[tid + s];
        if (ov > red_val[tid] || (ov == red_val[tid] && oo < red_oid[tid])) {
          red_val[tid] = ov;
          red_idx[tid] = oi;
          red_oid[tid] = oo;
        }
      }
      __syncthreads();
    }
    if (tid == 0) {
      const float bx = red_val[0];
      const int   bi = red_idx[0];
      const int   ok = (bx > NEG_INF) && (bi >= 0);
      kept_sc[it] = ok ? bx : NEG_INF;
      kept_id[it] = ok ? bi : -1;
      ok_flag     = ok;
      if (ok) {
        bbox[0] = x0a[bi]; bbox[1] = y0a[bi];
        bbox[2] = x1a[bi]; bbox[3] = y1a[bi];
        bbox[4] = ara[bi];
      }
    }
    __syncthreads();
    if (!ok_flag) {
      for (int j = it + tid; j < NMS_MAX_K; j += 256) {
        kept_sc[j] = NEG_INF;
        kept_id[j] = -1;
      }
      __syncthreads();
      break;
    }
    const float Bx0 = bbox[0], By0 = bbox[1], Bx1 = bbox[2], By1 = bbox[3];
    const float Ba  = bbox[4];
    for (int i = tid; i < cnt; i += 256) {
      float iw = fminf(x1a[i], Bx1) - fmaxf(x0a[i], Bx0);
      float ih = fminf(y1a[i], By1) - fmaxf(y0a[i], By0);
      iw = fmaxf(iw, 0.0f);
      ih = fmaxf(ih, 0.0f);
      const float inter = iw * ih;
      const float iou   = inter / (ara[i] + Ba - inter + 1e-12f);
      if (iou > IOU_THR) sc[i] = NEG_INF;   // suppresses the best box too (iou=1)
    }
    __syncthreads();
  }

  // kept_sc is non-increasing => top_k(.,10) == first 10 kept rows
  if (tid < NUM_PRED_K) {
    float* o = out + ((size_t)b * NUM_PRED_K + tid) * 6;
    const int   ki = kept_id[tid];
    const float ks = kept_sc[tid];
    if (ki >= 0) {
      o[0] = gbase[7 * CAP + ki];  // class id
      o[1] = ks;
      o[2] = x0a[ki];
      o[3] = y0a[ki];
      o[4] = x1a[ki];
      o[5] = y1a[ki];
    } else {
      o[0] = 0.0f; o[1] = 0.0f; o[2] = 0.0f;
      o[3] = 0.0f; o[4] = 0.0f; o[5] = 0.0f;
    }
  }
}

// ---------------- host entry ----------------
extern "C" void kernel_launch(void* const* d_in, const int* in_sizes, int n_in,
                              void* d_out, int out_size, void* d_ws, size_t ws_size,
                              hipStream_t stream) {
  (void)in_sizes; (void)n_in; (void)out_size; (void)ws_size;
  const float* y = (const float*)d_in[0];
  float* cand     = (float*)d_ws;
  int*   counters = (int*)((char*)d_ws + (size_t)BATCH * FIELDS * CAP * sizeof(float));

  hipMemsetAsync(counters, 0, BATCH * sizeof(int), stream);

  const int totalWaves = BATCH * N_BOXES_K;
  const int blocks = (totalWaves + 7) / 8;   // 8 waves (256 threads) per block
  ssd_decode_kernel<<<blocks, 256, 0, stream>>>(y, cand, counters);
  ssd_nms_kernel<<<BATCH, 256, 0, stream>>>(cand, counters, (float*)d_out);
}